// History_Selector_63651415327145
// MI455X (gfx1250) — compile-verified
//
#include <hip/hip_runtime.h>
#include <math.h>

typedef __attribute__((ext_vector_type(2))) float v2f;
typedef __attribute__((ext_vector_type(4))) float v4f;
typedef __attribute__((ext_vector_type(8))) float v8f;

#define B_      32
#define CDD_    5
#define HIS_    50
#define L_      32
#define LVL_    2
#define H_      256
#define K_      5
#define MROWS   55          // CDD_ + HIS_ rows projected per batch
#define PITCH   260         // LDS row pitch (260 % 64 banks == 4 -> conflict free)
#define TILE_ELEMS (L_*LVL_*H_)   // 16384 floats per gathered tile
#define THRESH  0.1f

// ---------------------------------------------------------------------------
// Kernel 1: per-batch  projection (WMMA f32 16x16x4) + L2-normalize + attn +
// top-k.  One block per batch, 8 waves.  Emits thresholded weights + indices.
// ---------------------------------------------------------------------------
__global__ __launch_bounds__(256)
void proj_attn_topk(const float* __restrict__ cdd, const float* __restrict__ his,
                    const float* __restrict__ W,   const float* __restrict__ bias,
                    float* __restrict__ ws_w, int* __restrict__ ws_idx)
{
    __shared__ float Y[MROWS * PITCH];     // projected rows (cdd then his)
    __shared__ float invn[MROWS];          // 1 / max(||row||, 1e-12)
    __shared__ float attn[CDD_ * HIS_];

    const int b   = blockIdx.x;
    const int tid = threadIdx.x;
    const int wv  = tid >> 5;        // wave id 0..7
    const int ln  = tid & 31;        // lane 0..31
    const int l16 = ln & 15;
    const int hs  = ln >> 4;         // half-wave select (K split for f32 WMMA)

    const float* cddB = cdd + (size_t)b * CDD_ * H_;
    const float* hisB = his + (size_t)b * HIS_ * H_;

    // ---- GEMM:  Y[55x256] = X @ W^T + bias  via v_wmma_f32_16x16x4_f32 ----
    // 4 m-tiles x 16 n-tiles = 64 output tiles, round-robin over 8 waves.
    for (int t = wv; t < 64; t += 8) {
        const int mt = t >> 4;               // 0..3
        const int nt = t & 15;               // 0..15
        const int rA = mt * 16 + l16;        // A row this lane feeds

        // A layout (32-bit 16x4): lanes 0-15 hold K={k0,k0+1}, lanes 16-31 K={k0+2,k0+3}
        float zm = 1.0f;
        const float* rowp;
        if      (rA < CDD_)  rowp = cddB + (size_t)rA * H_;
        else if (rA < MROWS) rowp = hisB + (size_t)(rA - CDD_) * H_;
        else               { rowp = cddB; zm = 0.0f; }

        const int n = nt * 16 + l16;         // B column (== output column)
        const float* wrow = W + (size_t)n * H_;   // B[k][n] = W[n][k]

        v8f acc = {};
        for (int k0 = 0; k0 < H_; k0 += 4) {
            const int kk = k0 + 2 * hs;
            v2f a = *(const v2f*)(rowp + kk);
            a.x *= zm; a.y *= zm;
            v2f bb = *(const v2f*)(wrow + kk);
            acc = __builtin_amdgcn_wmma_f32_16x16x4_f32(
                      false, a, false, bb, (short)0, acc, false, false);
        }

        // C/D layout: vgpr j -> row mt*16 + j + 8*hs, col n
        const float bv = bias[n];
        #pragma unroll
        for (int j = 0; j < 8; ++j) {
            const int row = mt * 16 + j + 8 * hs;
            if (row < MROWS) Y[row * PITCH + n] = acc[j] + bv;
        }
    }
    __syncthreads();

    // ---- row L2 norms (one wave per row, strided) ----
    for (int row = wv; row < MROWS; row += 8) {
        float s = 0.0f;
        #pragma unroll
        for (int i = 0; i < 8; ++i) {
            float v = Y[row * PITCH + ln + 32 * i];
            s += v * v;
        }
        for (int off = 16; off > 0; off >>= 1) s += __shfl_xor(s, off, 32);
        if (ln == 0) invn[row] = 1.0f / fmaxf(sqrtf(s), 1e-12f);
    }
    __syncthreads();

    // ---- attn[c][h] = <c_row, h_row> * invn_c * invn_h  (250 dots) ----
    for (int p = wv; p < CDD_ * HIS_; p += 8) {
        const int c = p / HIS_;
        const int h = p % HIS_;
        float s = 0.0f;
        #pragma unroll
        for (int i = 0; i < 8; ++i) {
            const int col = ln + 32 * i;
            s += Y[c * PITCH + col] * Y[(CDD_ + h) * PITCH + col];
        }
        for (int off = 16; off > 0; off >>= 1) s += __shfl_xor(s, off, 32);
        if (ln == 0) attn[p] = s * invn[c] * invn[CDD_ + h];
    }
    __syncthreads();

    // ---- top-k (descending, ties -> lowest index) + threshold ----
    if (tid < CDD_) {
        const int c = tid;
        float vals[HIS_];
        #pragma unroll
        for (int h = 0; h < HIS_; ++h) vals[h] = attn[c * HIS_ + h];
        for (int j = 0; j < K_; ++j) {
            float best = -__builtin_inff();
            int   bi   = 0;
            for (int h = 0; h < HIS_; ++h)
                if (vals[h] > best) { best = vals[h]; bi = h; }
            vals[bi] = -__builtin_inff();
            const int o = (b * CDD_ + c) * K_ + j;
            ws_idx[o] = bi;
            ws_w[o]   = (best < THRESH) ? 0.0f : best;
        }
    }
}

// ---------------------------------------------------------------------------
// Kernel 2: streaming gather + reweight.  One block per (b, c, j) tile:
// copies 16384 floats (64 KB) with 128-bit vectors (global_*_b128) and NT
// stores, plus the 32-wide gathered mask.
// ---------------------------------------------------------------------------
__global__ __launch_bounds__(256)
void gather_scale(const float* __restrict__ emb, const int* __restrict__ mask,
                  const float* __restrict__ ws_w, const int* __restrict__ ws_idx,
                  float* __restrict__ out)
{
    const int t   = blockIdx.x;              // 0..799 = ((b*CDD)+c)*K + j
    const int b   = t / (CDD_ * K_);
    const int tid = threadIdx.x;

    const int   hidx = ws_idx[t];
    const float wgt  = ws_w[t];

    const v4f* __restrict__ src =
        (const v4f*)(emb + ((size_t)b * HIS_ + hidx) * TILE_ELEMS);
    v4f* __restrict__ dst = (v4f*)(out + (size_t)t * TILE_ELEMS);

    #pragma unroll 4
    for (int i = 0; i < TILE_ELEMS / 4 / 256; ++i) {   // 16 iterations
        v4f v = src[tid + 256 * i];
        v *= wgt;
        __builtin_nontemporal_store(v, &dst[tid + 256 * i]);
    }

    if (tid < L_) {
        float* mout = out + (size_t)B_ * CDD_ * K_ * TILE_ELEMS + (size_t)t * L_;
        mout[tid] = (float)mask[((size_t)b * HIS_ + hidx) * L_ + tid];
    }
}

// ---------------------------------------------------------------------------
extern "C" void kernel_launch(void* const* d_in, const int* in_sizes, int n_in,
                              void* d_out, int out_size, void* d_ws, size_t ws_size,
                              hipStream_t stream)
{
    (void)in_sizes; (void)n_in; (void)out_size; (void)ws_size;

    const float* cdd  = (const float*)d_in[0];   // [32, 5, 256]
    const float* his  = (const float*)d_in[1];   // [32, 50, 256]
    const float* emb  = (const float*)d_in[2];   // [32, 50, 32, 2, 256]
    const int*   mask = (const int*)  d_in[3];   // [32, 50, 32]
    const float* W    = (const float*)d_in[4];   // [256, 256]
    const float* bias = (const float*)d_in[5];   // [256]

    float* out    = (float*)d_out;
    float* ws_w   = (float*)d_ws;                      // 800 floats
    int*   ws_idx = (int*)(ws_w + B_ * CDD_ * K_);     // 800 ints

    proj_attn_topk<<<dim3(B_), dim3(256), 0, stream>>>(cdd, his, W, bias,
                                                       ws_w, ws_idx);
    gather_scale<<<dim3(B_ * CDD_ * K_), dim3(256), 0, stream>>>(emb, mask,
                                                                 ws_w, ws_idx, out);
}